// RPN_3D_loss_81183471829161
// MI455X (gfx1250) — compile-verified
//
#include <hip/hip_runtime.h>
#include <stdint.h>
#include <math.h>

namespace {

constexpr int   kA  = 36;
constexpr int   kFH = 32;
constexpr int   kFW = 106;
constexpr int   kN  = kFH * kFW * kA;   // 122112 (divisible by 256: 477 blocks/batch)
constexpr int   kB  = 4;
constexpr int   kG  = 64;
constexpr float kFgThresh = 0.5f;
constexpr float kBgHi     = 0.5f;
constexpr float kBgLo     = 0.0f;

__device__ __forceinline__ float smooth_l1(float x) {
  float ax = fabsf(x);
  return ax < 1.0f ? 0.5f * ax * ax : ax - 0.5f;
}

// CDNA5 async global->LDS copy (ASYNCcnt-tracked). VDST carries the LDS byte
// address, VADDR the 64-bit global address (GV mode), per ISA 15.18.3 op 96.
__device__ __forceinline__ void async_ld_b32(uint32_t lds_off, const void* gaddr) {
  asm volatile("global_load_async_to_lds_b32 %0, %1, off"
               :: "v"(lds_off), "v"(gaddr)
               : "memory");
}
__device__ __forceinline__ void wait_async_zero() {
  asm volatile("s_wait_asynccnt 0" ::: "memory");
}

// wave32 shuffle reduce + cross-wave LDS combine for 5 accumulators.
// Result valid in v[] on thread 0.
__device__ __forceinline__ void block_reduce5(float v[5]) {
  __shared__ float buf[8][5];
  const int lane = threadIdx.x & 31;
  const int wave = threadIdx.x >> 5;
#pragma unroll
  for (int k = 0; k < 5; ++k) {
    float x = v[k];
#pragma unroll
    for (int off = 16; off > 0; off >>= 1) x += __shfl_down(x, off, 32);
    v[k] = x;
  }
  if (lane == 0) {
#pragma unroll
    for (int k = 0; k < 5; ++k) buf[wave][k] = v[k];
  }
  __syncthreads();
  if (threadIdx.x == 0) {
#pragma unroll
    for (int k = 0; k < 5; ++k) {
      float s = 0.f;
#pragma unroll
      for (int w = 0; w < 8; ++w) s += buf[w][k];
      v[k] = s;
    }
  }
}

__global__ __launch_bounds__(256) void rpn_loss_main(
    const float* __restrict__ cls,
    const float* __restrict__ bbox2d,
    const float* __restrict__ bbox3d,
    const float* __restrict__ rois,
    const float* __restrict__ anchors,
    const float* __restrict__ means,
    const float* __restrict__ stds,
    const float* __restrict__ gt_boxes,
    const float* __restrict__ gt3d,
    const int*   __restrict__ gt_labels,
    float* __restrict__ partials) {
  __shared__ alignas(16) float s_gt[kG * 4];  // gt boxes (x1,y1,x2,y2)
  __shared__ float s_g3[kG * 7];              // gt boxes 3d
  __shared__ int   s_lb[kG];                  // gt labels

  const int tid = threadIdx.x;
  const int gi  = blockIdx.x * 256 + tid;
  const int b   = gi / kN;          // whole block lies in one batch
  const int n   = gi - b * kN;

  // ---- stage this batch's GT tensors into LDS with async copies ----
  {
    const uint32_t l0 = (uint32_t)(uintptr_t)&s_gt[0];
    const uint32_t l1 = (uint32_t)(uintptr_t)&s_g3[0];
    const uint32_t l2 = (uint32_t)(uintptr_t)&s_lb[0];
    const float* g0 = gt_boxes  + (size_t)b * (kG * 4);
    const float* g1 = gt3d      + (size_t)b * (kG * 7);
    const int*   g2 = gt_labels + (size_t)b * kG;
    // 256 + 448 + 64 = 768 dwords = 3 per thread
#pragma unroll
    for (int r = 0; r < 3; ++r) {
      const int d = tid + r * 256;
      if (d < kG * 4) {
        async_ld_b32(l0 + (uint32_t)d * 4u, g0 + d);
      } else if (d < kG * 4 + kG * 7) {
        const int j = d - kG * 4;
        async_ld_b32(l1 + (uint32_t)j * 4u, g1 + j);
      } else {
        const int j = d - (kG * 4 + kG * 7);
        async_ld_b32(l2 + (uint32_t)j * 4u, g2 + j);
      }
    }
  }
  wait_async_zero();
  __syncthreads();

  // ---- per-ROI geometry ----
  const float rx1 = rois[(size_t)n * 5 + 0];
  const float ry1 = rois[(size_t)n * 5 + 1];
  const float rx2 = rois[(size_t)n * 5 + 2];
  const float ry2 = rois[(size_t)n * 5 + 3];
  const int  aidx = (int)rois[(size_t)n * 5 + 4];
  const float w  = rx2 - rx1 + 1.0f;
  const float h  = ry2 - ry1 + 1.0f;
  const float cx = rx1 + 0.5f * w;
  const float cy = ry1 + 0.5f * h;
  const float area_r = w * h;

  // ---- IoU max/argmax over G (first-max tie-break, like jnp.argmax) ----
  float best = -INFINITY;
  int   bg_i = 0;
#pragma unroll 4
  for (int g = 0; g < kG; ++g) {
    const float4 gb = ((const float4*)s_gt)[g];
    float iw = fminf(rx2, gb.z) - fmaxf(rx1, gb.x) + 1.0f;
    float ih = fminf(ry2, gb.w) - fmaxf(ry1, gb.y) + 1.0f;
    iw = fmaxf(iw, 0.0f);
    ih = fmaxf(ih, 0.0f);
    const float inter  = iw * ih;
    const float area_g = (gb.z - gb.x + 1.0f) * (gb.w - gb.y + 1.0f);
    const float iou = inter / (area_r + area_g - inter);
    if (iou > best) { best = iou; bg_i = g; }
  }

  const bool fg = best >= kFgThresh;
  const bool bg = (!fg) && (best < kBgHi) && (best >= kBgLo);
  const float fgf  = fg ? 1.0f : 0.0f;
  const float wcls = (fg || bg) ? 1.0f : 0.0f;

  // ---- 2D regression target + smooth L1 ----
  const float4 gbb = ((const float4*)s_gt)[bg_i];
  const float gw  = gbb.z - gbb.x + 1.0f;
  const float gh  = gbb.w - gbb.y + 1.0f;
  const float gcx = gbb.x + 0.5f * gw;
  const float gcy = gbb.y + 0.5f * gh;
  float t2[4];
  t2[0] = (gcx - cx) / w;
  t2[1] = (gcy - cy) / h;
  t2[2] = logf(gw / w);
  t2[3] = logf(gh / h);
  float l2d = 0.0f;
  const float* bb2 = bbox2d + ((size_t)b * kN + n) * 4;
#pragma unroll
  for (int k = 0; k < 4; ++k) {
    const float t = (t2[k] - means[k]) / stds[k];
    l2d += smooth_l1(bb2[k] - t);
  }

  // ---- 3D regression target + smooth L1 ----
  const float* a9 = anchors + (size_t)aidx * 9;
  const float* g3 = &s_g3[bg_i * 7];
  float t3[7];
  t3[0] = (g3[0] - cx) / w;
  t3[1] = (g3[1] - cy) / h;
  t3[2] = g3[2] - a9[4];
  t3[3] = logf(g3[3] / a9[5]);
  t3[4] = logf(g3[4] / a9[6]);
  t3[5] = logf(g3[5] / a9[7]);
  t3[6] = g3[6] - a9[8];
  float l3d = 0.0f;
  const float* bb3 = bbox3d + ((size_t)b * kN + n) * 7;
#pragma unroll
  for (int k = 0; k < 7; ++k) {
    const float t = (t3[k] - means[4 + k]) / stds[4 + k];
    l3d += smooth_l1(bb3[k] - t);
  }

  // ---- cross entropy over NC=4 ----
  const float* c = cls + ((size_t)b * kN + n) * 4;
  const float c0 = c[0], c1 = c[1], c2 = c[2], c3 = c[3];
  const float m  = fmaxf(fmaxf(c0, c1), fmaxf(c2, c3));
  const float se = expf(c0 - m) + expf(c1 - m) + expf(c2 - m) + expf(c3 - m);
  const float lse = m + logf(se);
  const int lbl = fg ? s_lb[bg_i] : 0;
  const float cl = (lbl == 0) ? c0 : (lbl == 1) ? c1 : (lbl == 2) ? c2 : c3;
  const float ce = lse - cl;

  // ---- block reduction of the 5 global accumulators ----
  float acc[5];
  acc[0] = fgf;          // n_fg
  acc[1] = wcls;         // n_cls
  acc[2] = fgf * l2d;    // sum 2d loss
  acc[3] = fgf * l3d;    // sum 3d loss
  acc[4] = wcls * ce;    // sum ce
  block_reduce5(acc);
  if (tid == 0) {
#pragma unroll
    for (int k = 0; k < 5; ++k) partials[(size_t)blockIdx.x * 5 + k] = acc[k];
  }
}

__global__ __launch_bounds__(256) void rpn_loss_finalize(
    const float* __restrict__ partials, int nblocks, float* __restrict__ out) {
  float acc[5] = {0.f, 0.f, 0.f, 0.f, 0.f};
  for (int i = threadIdx.x; i < nblocks; i += 256) {
#pragma unroll
    for (int k = 0; k < 5; ++k) acc[k] += partials[(size_t)i * 5 + k];
  }
  block_reduce5(acc);
  if (threadIdx.x == 0) {
    const float n_fg  = fmaxf(acc[0], 1.0f);
    const float n_cls = fmaxf(acc[1], 1.0f);
    out[0] = acc[4] / n_cls + acc[2] / n_fg + acc[3] / n_fg;
  }
}

} // namespace

extern "C" void kernel_launch(void* const* d_in, const int* in_sizes, int n_in,
                              void* d_out, int out_size, void* d_ws, size_t ws_size,
                              hipStream_t stream) {
  (void)in_sizes; (void)n_in; (void)out_size; (void)ws_size;
  const float* cls     = (const float*)d_in[0];
  // d_in[1] = prob : unused by the reference loss
  const float* bbox2d  = (const float*)d_in[2];
  const float* bbox3d  = (const float*)d_in[3];
  const float* rois    = (const float*)d_in[4];
  const float* anchors = (const float*)d_in[5];
  const float* means   = (const float*)d_in[6];
  const float* stds    = (const float*)d_in[7];
  const float* gtb     = (const float*)d_in[8];
  const float* gt3     = (const float*)d_in[9];
  const int*   gtl     = (const int*)d_in[10];
  float* out      = (float*)d_out;
  float* partials = (float*)d_ws;   // nblocks * 5 floats

  const int total   = kB * kN;              // 488448
  const int nblocks = (total + 255) / 256;  // 1908 (exact)

  rpn_loss_main<<<nblocks, 256, 0, stream>>>(cls, bbox2d, bbox3d, rois, anchors,
                                             means, stds, gtb, gt3, gtl, partials);
  rpn_loss_finalize<<<1, 256, 0, stream>>>(partials, nblocks, out);
}